// DocumentGAT_11785390260817
// MI455X (gfx1250) — compile-verified
//
#include <hip/hip_runtime.h>
#include <hip/hip_bf16.h>
#include <math.h>

typedef __attribute__((ext_vector_type(2))) float v2f;
typedef __attribute__((ext_vector_type(8))) float v8f;

// ---------------------------------------------------------------------------
// CDNA5 async-copy helpers (memory -> LDS, tracked by ASYNCcnt)
// ---------------------------------------------------------------------------
__device__ __forceinline__ unsigned int lds_off32(const void* p) {
    // LDS aperture occupies ADDR[63:32]; low 32 bits of a generic pointer to
    // LDS are the raw LDS byte offset (ISA 10.2 aperture calc).
    return (unsigned int)(uintptr_t)p;
}

__device__ __forceinline__ void async_copy16(unsigned int lds, const float* gp) {
    asm volatile("global_load_async_to_lds_b128 %0, %1, off"
                 :: "v"(lds), "v"(gp) : "memory");
}

__device__ __forceinline__ void wait_async_le2() {
    asm volatile("s_wait_asynccnt 0x2" ::: "memory");
}
__device__ __forceinline__ void wait_async_0() {
    asm volatile("s_wait_asynccnt 0x0" ::: "memory");
}

// ---------------------------------------------------------------------------
// fp32 WMMA GEMM: C[M,N] = A[M,K] @ B[K,N] (+ bias[N] if bias != nullptr)
// Requirements: K % 8 == 0, N % 64 == 0 (true for all uses: K=5000/256, N=256)
// Block: 256 threads (8 wave32). Block tile 128x64, wave tile 32x32.
// Double-buffered LDS fed by global_load_async_to_lds_b128; per batch each
// wave issues exactly 2 async instructions, so s_wait_asynccnt 2 retires the
// previous batch while the next streams in (async loads complete in order).
// ---------------------------------------------------------------------------
__global__ __launch_bounds__(256) void gemm_f32_wmma(
    const float* __restrict__ A, const float* __restrict__ B,
    const float* __restrict__ bias, float* __restrict__ C,
    int M, int K, int N)
{
    __shared__ float Abuf[2][128 * 9];   // 128 rows x 8 K, padded stride 9
    __shared__ float Bbuf[2][8 * 68];    // 8 K x 64 cols, padded stride 68

    const int tid  = threadIdx.x;
    const int lane = tid & 31;
    const int wave = tid >> 5;
    const int wr   = wave >> 1;          // 0..3 wave row
    const int wc   = wave & 1;           // 0..1 wave col
    const int bm   = blockIdx.y * 128;
    const int bn   = blockIdx.x * 64;

    const int lrow  = lane & 15;         // M / N index within 16
    const int khalf = (lane >> 4) << 1;  // K sub-offset: 0 or 2

    // --- A staging: thread t copies 16B of row (t>>1), K-chunk (t&1)*4 ---
    const int arow = tid >> 1;
    const int acol = (tid & 1) << 2;
    const int garw = (bm + arow < M) ? (bm + arow) : (M - 1); // clamp, keep EXEC!=0
    const float* aglob = A + (size_t)garw * K + acol;         // + k0 at issue
    const unsigned int alds[2] = { lds_off32(&Abuf[0][arow * 9 + acol]),
                                   lds_off32(&Abuf[1][arow * 9 + acol]) };

    // --- B staging: wave w, lanes 0..15 copy row w, 16B chunk lane*4 ---
    const float* bglob = B + (size_t)wave * N + bn + (lane << 2); // + k0*N at issue
    const unsigned int blds[2] = { lds_off32(&Bbuf[0][wave * 68 + (lane << 2)]),
                                   lds_off32(&Bbuf[1][wave * 68 + (lane << 2)]) };

    v8f acc[2][2] = {};

    auto issue_batch = [&](int bi, int k0) {
        async_copy16(alds[bi], aglob + k0);                 // 1 instr / wave
        if (lane < 16)                                      // EXEC=0xFFFF != 0
            async_copy16(blds[bi], bglob + (size_t)k0 * N); // 1 instr / wave
    };

    issue_batch(0, 0);
    const int nk = K >> 3;
    for (int kt = 0; kt < nk; ++kt) {
        const int cur = kt & 1;
        if (kt + 1 < nk) {
            issue_batch(cur ^ 1, (kt + 1) << 3);
            wait_async_le2();        // previous batch (this buffer) complete
        } else {
            wait_async_0();
        }
        __syncthreads();             // all waves' async data visible

        const float* Ads = Abuf[cur];
        const float* Bds = Bbuf[cur];
        #pragma unroll
        for (int kk = 0; kk < 8; kk += 4) {
            v2f afrag[2], bfrag[2];
            #pragma unroll
            for (int i = 0; i < 2; ++i) {
                int r = wr * 32 + i * 16 + lrow;
                afrag[i].x = Ads[r * 9 + kk + khalf + 0];
                afrag[i].y = Ads[r * 9 + kk + khalf + 1];
            }
            #pragma unroll
            for (int j = 0; j < 2; ++j) {
                int c = wc * 32 + j * 16 + lrow;
                bfrag[j].x = Bds[(kk + khalf + 0) * 68 + c];
                bfrag[j].y = Bds[(kk + khalf + 1) * 68 + c];
            }
            #pragma unroll
            for (int i = 0; i < 2; ++i)
                #pragma unroll
                for (int j = 0; j < 2; ++j)
                    acc[i][j] = __builtin_amdgcn_wmma_f32_16x16x4_f32(
                        false, afrag[i], false, bfrag[j],
                        (short)0, acc[i][j], false, false);
        }
        __syncthreads();             // done reading before buffer is reused
    }

    // Store: C/D layout — lane l, vgpr r -> row r + 8*(l>>4), col l&15.
    #pragma unroll
    for (int i = 0; i < 2; ++i) {
        #pragma unroll
        for (int j = 0; j < 2; ++j) {
            int col = bn + wc * 32 + j * 16 + lrow;
            #pragma unroll
            for (int r = 0; r < 8; ++r) {
                int row = bm + wr * 32 + i * 16 + r + ((lane >> 4) << 3);
                if (row < M) {
                    float v = acc[i][j][r];
                    if (bias) v += bias[col];
                    C[(size_t)row * N + col] = v;
                }
            }
        }
    }
}

// ---------------------------------------------------------------------------
// Small helpers
// ---------------------------------------------------------------------------
__global__ void fill_kernel(float* __restrict__ p, float v, size_t n) {
    size_t i = (size_t)blockIdx.x * blockDim.x + threadIdx.x;
    if (i < n) p[i] = v;
}

__device__ inline void atomicMaxF(float* addr, float val) {
    // exact (associative) float max via int/uint ordering trick
    if (val >= 0.f) atomicMax((int*)addr, __float_as_int(val));
    else            atomicMin((unsigned int*)addr, __float_as_uint(val));
}

// s_src/s_dst: per-(node,head) dot of features with attention vectors
__global__ void gat_scores(const float* __restrict__ hf,
                           const float* __restrict__ asrc,
                           const float* __restrict__ adst,
                           float* __restrict__ ssrc, float* __restrict__ sdst,
                           int N, int H, int C) {
    int t = blockIdx.x * blockDim.x + threadIdx.x;
    if (t >= N * H) return;
    int n = t / H, h = t % H;
    const float* hp = hf + (size_t)n * 256 + h * C;
    const float* ap = asrc + h * C;
    const float* bp = adst + h * C;
    float s0 = 0.f, s1 = 0.f;
    for (int c = 0; c < C; ++c) { float v = hp[c]; s0 += v * ap[c]; s1 += v * bp[c]; }
    ssrc[t] = s0; sdst[t] = s1;
}

// pass 1: e = leaky_relu(s_src[src]+s_dst[dst]); segment max into m[dst]
__global__ void edge_max(const int* __restrict__ ei, int E, int N, int H,
                         const float* __restrict__ ssrc, const float* __restrict__ sdst,
                         float* __restrict__ ebuf, float* __restrict__ m) {
    int e = blockIdx.x * blockDim.x + threadIdx.x;
    int Et = E + N;
    if (e >= Et) return;
    int s, d;
    if (e < E) { s = ei[e]; d = ei[E + e]; } else { s = e - E; d = s; }
    for (int h = 0; h < H; ++h) {
        float v = ssrc[s * H + h] + sdst[d * H + h];
        v = (v >= 0.f) ? v : 0.2f * v;            // leaky_relu 0.2
        ebuf[(size_t)e * H + h] = v;
        atomicMaxF(&m[d * H + h], v);
    }
}

// pass 2: p = exp(e - m[dst]); segment sum into z[dst]
__global__ void edge_exp_sum(const int* __restrict__ ei, int E, int N, int H,
                             const float* __restrict__ m, float* __restrict__ ebuf,
                             float* __restrict__ z) {
    size_t t = (size_t)blockIdx.x * blockDim.x + threadIdx.x;
    size_t tot = (size_t)(E + N) * H;
    if (t >= tot) return;
    int e = (int)(t / H), h = (int)(t % H);
    int d = (e < E) ? ei[E + e] : e - E;
    float p = expf(ebuf[t] - m[d * H + h]);
    ebuf[t] = p;
    atomicAdd(&z[d * H + h], p);
}

// pass 3: out[dst] += (p/z[dst]) * h[src]   (block = 1 edge, thread = feature)
__global__ __launch_bounds__(256) void edge_scatter(
    const int* __restrict__ ei, int E, int N, int H, int logC,
    const float* __restrict__ ebuf, const float* __restrict__ z,
    const float* __restrict__ hf, float* __restrict__ out) {
    int e = blockIdx.x;
    int d = threadIdx.x;
    int s, dn;
    if (e < E) { s = ei[e]; dn = ei[E + e]; } else { s = e - E; dn = s; }
    int h = d >> logC;
    float alpha = ebuf[(size_t)e * H + h] / z[dn * H + h];
    atomicAdd(&out[(size_t)dn * 256 + d], alpha * hf[(size_t)s * 256 + d]);
}

// bias + ELU + residual + LayerNorm(256) : one block per node
__global__ __launch_bounds__(256) void gat_post(
    const float* __restrict__ agg, const float* __restrict__ bias,
    const float* __restrict__ xin, const float* __restrict__ g,
    const float* __restrict__ be, float* __restrict__ xout) {
    int n = blockIdx.x;
    int d = threadIdx.x;
    float v = agg[(size_t)n * 256 + d] + bias[d];
    v = (v > 0.f) ? v : expm1f(v);               // ELU(alpha=1)
    v += xin[(size_t)n * 256 + d];

    __shared__ float red[8];
    float s = v;
    #pragma unroll
    for (int o = 16; o > 0; o >>= 1) s += __shfl_down(s, o, 32);
    if ((d & 31) == 0) red[d >> 5] = s;
    __syncthreads();
    float tot = 0.f;
    #pragma unroll
    for (int i = 0; i < 8; ++i) tot += red[i];
    float mu = tot * (1.0f / 256.0f);
    float dv = v - mu;
    __syncthreads();

    float s2 = dv * dv;
    #pragma unroll
    for (int o = 16; o > 0; o >>= 1) s2 += __shfl_down(s2, o, 32);
    if ((d & 31) == 0) red[d >> 5] = s2;
    __syncthreads();
    float tot2 = 0.f;
    #pragma unroll
    for (int i = 0; i < 8; ++i) tot2 += red[i];
    float var = tot2 * (1.0f / 256.0f);

    xout[(size_t)n * 256 + d] = dv * rsqrtf(var + 1e-5f) * g[d] + be[d];
}

// mean-pool accumulation (sums + counts)
__global__ void pool_kernel(const float* __restrict__ x2, const float* __restrict__ x0,
                            const int* __restrict__ batch,
                            float* __restrict__ xg, float* __restrict__ xs,
                            float* __restrict__ cnt, int N) {
    size_t t = (size_t)blockIdx.x * blockDim.x + threadIdx.x;
    if (t >= (size_t)N * 256) return;
    int n = (int)(t >> 8), d = (int)(t & 255);
    int b = batch[n];
    atomicAdd(&xg[b * 256 + d], x2[t]);
    atomicAdd(&xs[b * 256 + d], x0[t]);
    if (d == 0) atomicAdd(&cnt[b], 1.0f);
}

// MLP head + log_softmax : one block (256 thr) per batch element
__global__ __launch_bounds__(256) void head_kernel(
    const float* __restrict__ xg, const float* __restrict__ xs,
    const float* __restrict__ cnt,
    const float* __restrict__ Wf,  const float* __restrict__ bf,
    const float* __restrict__ Wc1, const float* __restrict__ bc1,
    const float* __restrict__ Wc2, const float* __restrict__ bc2,
    float* __restrict__ out) {
    int b = blockIdx.x;
    int t = threadIdx.x;
    __shared__ float xc[512];
    __shared__ float f[256];
    __shared__ float c1[128];
    __shared__ float logits[20];

    float inv = 1.f / fmaxf(cnt[b], 1.f);
    xc[t]       = xg[b * 256 + t] * inv;
    xc[256 + t] = xs[b * 256 + t] * inv;
    __syncthreads();

    float acc = bf[t];
    for (int i = 0; i < 512; ++i) acc += xc[i] * Wf[i * 256 + t];
    f[t] = fmaxf(acc, 0.f);
    __syncthreads();

    if (t < 128) {
        float a = bc1[t];
        for (int i = 0; i < 256; ++i) a += f[i] * Wc1[i * 128 + t];
        c1[t] = fmaxf(a, 0.f);
    }
    __syncthreads();

    if (t < 20) {
        float a = bc2[t];
        for (int i = 0; i < 128; ++i) a += c1[i] * Wc2[i * 20 + t];
        logits[t] = a;
    }
    __syncthreads();

    if (t < 20) {
        float mx = logits[0];
        for (int i = 1; i < 20; ++i) mx = fmaxf(mx, logits[i]);
        float se = 0.f;
        for (int i = 0; i < 20; ++i) se += expf(logits[i] - mx);
        out[b * 20 + t] = logits[t] - mx - logf(se);
    }
}

// ---------------------------------------------------------------------------
// Host launcher
// ---------------------------------------------------------------------------
extern "C" void kernel_launch(void* const* d_in, const int* in_sizes, int n_in,
                              void* d_out, int out_size, void* d_ws, size_t ws_size,
                              hipStream_t stream) {
    const float* x    = (const float*)d_in[0];
    const int*   ei   = (const int*)  d_in[1];
    const int*   batch= (const int*)  d_in[2];
    const float* W_in = (const float*)d_in[3];
    const float* b_in = (const float*)d_in[4];
    const float* W1   = (const float*)d_in[5];
    const float* as1  = (const float*)d_in[6];
    const float* ad1  = (const float*)d_in[7];
    const float* b1   = (const float*)d_in[8];
    const float* g1   = (const float*)d_in[9];
    const float* be1  = (const float*)d_in[10];
    const float* W2   = (const float*)d_in[11];
    const float* as2  = (const float*)d_in[12];
    const float* ad2  = (const float*)d_in[13];
    const float* b2   = (const float*)d_in[14];
    const float* g2   = (const float*)d_in[15];
    const float* be2  = (const float*)d_in[16];
    const float* Wf   = (const float*)d_in[17];
    const float* bf   = (const float*)d_in[18];
    const float* Wc1  = (const float*)d_in[19];
    const float* bc1  = (const float*)d_in[20];
    const float* Wc2  = (const float*)d_in[21];
    const float* bc2  = (const float*)d_in[22];

    const int N = in_sizes[2];            // 40000 nodes
    const int E = in_sizes[1] / 2;        // 1.28M edges
    const int V = in_sizes[0] / N;        // 5000
    const int D = 256;
    const int B = 64;

    // workspace layout (floats)
    float* w = (float*)d_ws;
    const size_t ND = (size_t)N * D;
    float* x0   = w;                 w += ND;
    float* x1   = w;                 w += ND;
    float* x2   = w;                 w += ND;
    float* hf   = w;                 w += ND;
    float* agg  = w;                 w += ND;
    float* ssrc = w;                 w += (size_t)N * 8;
    float* sdst = w;                 w += (size_t)N * 8;
    float* mbuf = w;                 w += (size_t)N * 8;
    float* zbuf = w;                 w += (size_t)N * 8;
    float* ebuf = w;                 w += (size_t)(E + N) * 8;
    float* pxg  = w;                 w += (size_t)B * D;
    float* pxs  = w;                 w += (size_t)B * D;
    float* cnt  = w;                 w += B;

    const float NEG_INF = -__builtin_huge_valf();

    auto gemm = [&](const float* A, const float* Bm, const float* bias,
                    float* Cm, int M, int K, int Nn) {
        dim3 grid(Nn / 64, (M + 127) / 128);
        gemm_f32_wmma<<<grid, 256, 0, stream>>>(A, Bm, bias, Cm, M, K, Nn);
    };

    // ---- input projection: x0 = x @ W_in + b_in (the 102-GFLOP GEMM) ----
    gemm(x, W_in, b_in, x0, N, V, D);

    auto run_layer = [&](const float* xin, float* xout,
                         const float* W, const float* asrc, const float* adst,
                         const float* bias, const float* g, const float* be,
                         int H, int C, int logC) {
        gemm(xin, W, nullptr, hf, N, D, D);                       // h = x @ W
        int NH = N * H;
        gat_scores<<<(NH + 255) / 256, 256, 0, stream>>>(hf, asrc, adst,
                                                         ssrc, sdst, N, H, C);
        fill_kernel<<<(NH + 255) / 256, 256, 0, stream>>>(mbuf, NEG_INF, (size_t)NH);
        fill_kernel<<<(NH + 255) / 256, 256, 0, stream>>>(zbuf, 0.f, (size_t)NH);
        fill_kernel<<<(int)((ND + 255) / 256), 256, 0, stream>>>(agg, 0.f, ND);

        int Et = E + N;
        edge_max<<<(Et + 255) / 256, 256, 0, stream>>>(ei, E, N, H,
                                                       ssrc, sdst, ebuf, mbuf);
        size_t EtH = (size_t)Et * H;
        edge_exp_sum<<<(int)((EtH + 255) / 256), 256, 0, stream>>>(ei, E, N, H,
                                                                   mbuf, ebuf, zbuf);
        edge_scatter<<<Et, 256, 0, stream>>>(ei, E, N, H, logC,
                                             ebuf, zbuf, hf, agg);
        gat_post<<<N, 256, 0, stream>>>(agg, bias, xin, g, be, xout);
    };

    run_layer(x0, x1, W1, as1, ad1, b1, g1, be1, 8, 32, 5);   // H1=8,  C1=32
    run_layer(x1, x2, W2, as2, ad2, b2, g2, be2, 4, 64, 6);   // H2=4,  C2=64

    // ---- pooling + head ----
    fill_kernel<<<(B * D * 2 + B + 255) / 256, 256, 0, stream>>>(pxg, 0.f,
        (size_t)(B * D * 2 + B));   // zeros pxg, pxs, cnt (contiguous)
    pool_kernel<<<(int)((ND + 255) / 256), 256, 0, stream>>>(x2, x0, batch,
                                                             pxg, pxs, cnt, N);
    head_kernel<<<B, 256, 0, stream>>>(pxg, pxs, cnt, Wf, bf, Wc1, bc1,
                                       Wc2, bc2, (float*)d_out);
}